// RNN_83013127897849
// MI455X (gfx1250) — compile-verified
//
#include <hip/hip_runtime.h>
#include <math.h>

typedef float v2f __attribute__((ext_vector_type(2)));
typedef float v4f __attribute__((ext_vector_type(4)));
typedef float v8f __attribute__((ext_vector_type(8)));

#if defined(__has_builtin)
#if __has_builtin(__builtin_amdgcn_tanhf)
#define TANHF(v) __builtin_amdgcn_tanhf(v)
#else
#define TANHF(v) tanhf(v)
#endif
#else
#define TANHF(v) tanhf(v)
#endif

#define HID 32
// DS_SWIZZLE group-of-32 SWAPX16 (and=0x1f, or=0, xor=0x10) - used only for the
// output-projection cross-half reduce (off the recurrence critical path).
#define SWAPX16_IMM 0x401F

#define WMMA(Aop, Bop, Cop) \
  __builtin_amdgcn_wmma_f32_16x16x4_f32(false, (Aop), false, (Bop), (short)0, (Cop), false, false)

__device__ __forceinline__ float swapx16(float v) {
  return __int_as_float(__builtin_amdgcn_ds_swizzle(__float_as_int(v), SWAPX16_IMM));
}

// Hidden-unit relabeling: unit u sits at K-position sigma(u) in the A/B operands,
//   sigma(16T + 8h + 2c + v) = 16T + 4c + 2h + v
// chosen so the WMMA D-output register layout IS the next step's B-operand layout
// (zero-cost transpose). kinv is sigma^-1 (K-position -> hidden unit).
__device__ __forceinline__ int kinv(int k) {
  const int T = (k >> 4) & 1, kk = k & 15;
  return T * 16 + ((kk >> 1) & 1) * 8 + (kk >> 2) * 2 + (kk & 1);
}

// One wave32 per block; each wave owns a 16-batch tile and runs the whole T-scan.
// No LDS arrays, no barriers, no per-step shuffles on the recurrence path.
// Per step (transposed formulation): D(32x16) = A(32x36) x B(36x16), with
//   A = [W_hh | W_ih | b_ih+b_hh | 0 | 0]  (constant, K-permuted by sigma, in VGPRs)
//   B = [h_{t-1} | x_t | 1 | 0 | 0]^T      (dynamic; D of step t-1, renamed)
__global__ __launch_bounds__(32)
void rnn_scan_kernel(const float* __restrict__ x,     // [B,T,1]
                     const float* __restrict__ h0,    // [1,B,32]
                     const float* __restrict__ W_ih,  // [32,1]
                     const float* __restrict__ b_ih,  // [32]
                     const float* __restrict__ W_hh,  // [32,32]
                     const float* __restrict__ b_hh,  // [32]
                     const float* __restrict__ W_out, // [1,32]
                     const float* __restrict__ b_out, // [1]
                     float* __restrict__ out,         // [B*T] outs then [B*32] hT
                     int Bsz, int T)
{
  const int  lane = threadIdx.x & 31;
  const int  L    = lane & 15;
  const bool up   = lane >= 16;
  const int  kap  = up ? 2 : 0;        // K sub-offset within a 4-chunk (A/B layouts)
  const int  h8   = up ? 8 : 0;        // hidden-unit offset owned by this lane half
  const int  base = blockIdx.x * 16;
  const int  b    = base + L;          // this lane's batch row (lanes n, n+16 share it)

  // ---- constant A operands: A[tile][c] = W_hh rows j=tile*16+L, K-cols via sigma^-1 ----
  v2f A0[9], A1[9];
  #pragma unroll
  for (int c = 0; c < 9; ++c) {
    float a00, a01, a10, a11;
    if (c < 8) {
      const int u0 = kinv(4 * c + kap), u1 = kinv(4 * c + kap + 1);
      a00 = W_hh[L * HID + u0];
      a01 = W_hh[L * HID + u1];
      a10 = W_hh[(16 + L) * HID + u0];
      a11 = W_hh[(16 + L) * HID + u1];
    } else if (!up) {                   // K rows 32 (W_ih) and 33 (combined bias)
      a00 = W_ih[L];      a01 = b_ih[L] + b_hh[L];
      a10 = W_ih[16 + L]; a11 = b_ih[16 + L] + b_hh[16 + L];
    } else {                            // K rows 34,35 = zero padding
      a00 = a01 = a10 = a11 = 0.f;
    }
    A0[c] = (v2f){a00, a01};
    A1[c] = (v2f){a10, a11};
  }

  // W_out slices matching the D layout (lane half owns units h8+r and 16+h8+r)
  float woA[8], woB[8];
  #pragma unroll
  for (int r = 0; r < 8; ++r) {
    woA[r] = W_out[h8 + r];
    woB[r] = W_out[16 + h8 + r];
  }
  const float bo = b_out[0];

  // ---- dynamic B chunks from h0: bc[c] holds units off(c)..off(c)+1 of batch b ----
  // off(c) = (c<4 ? 2c : 16+2(c-4)) + h8   (contiguous pairs -> b64 loads)
  v2f bc[8];
  {
    const float* h0p = h0 + (size_t)b * HID;
    #pragma unroll
    for (int c = 0; c < 8; ++c) {
      const int off = (c < 4 ? 2 * c : 16 + 2 * (c - 4)) + h8;
      bc[c] = *reinterpret_cast<const v2f*>(&h0p[off]);
    }
  }

  const float* xrow   = x   + (size_t)b * T;
  float*       outrow = out + (size_t)b * T;

  for (int t4 = 0; t4 < T; t4 += 4) {
    const v4f xq = *reinterpret_cast<const v4f*>(&xrow[t4]);  // 4 timesteps of x
    const int pf = (t4 + 16 < T) ? (t4 + 16) : (T - 1);
    __builtin_prefetch(&xrow[pf], 0, 0);                      // warm next x cacheline
    v4f outv;
    #pragma unroll
    for (int u = 0; u < 4; ++u) {
      // B chunk 8: [x_t, 1 | 0, 0]
      const v2f b8 = up ? (v2f){0.f, 0.f} : (v2f){xq[u], 1.f};

      // ---- 18 WMMAs as 4 interleaved accumulation chains (depth 4/5) ----
      const v8f z = (v8f){0.f, 0.f, 0.f, 0.f, 0.f, 0.f, 0.f, 0.f};
      v8f d0a = z, d0b = z, d1a = z, d1b = z;
      #pragma unroll
      for (int c = 0; c < 8; c += 2) {
        d0a = WMMA(A0[c],     bc[c],     d0a);
        d1a = WMMA(A1[c],     bc[c],     d1a);
        d0b = WMMA(A0[c + 1], bc[c + 1], d0b);
        d1b = WMMA(A1[c + 1], bc[c + 1], d1b);
      }
      d0b = WMMA(A0[8], b8, d0b);
      d1b = WMMA(A1[8], b8, d1b);
      const v8f d0 = d0a + d0b;        // tile0: units h8+0..7       (per lane half)
      const v8f d1 = d1a + d1b;        // tile1: units 16+h8+0..7

      // ---- tanh -> h_t, fused Linear(32->1) partials ----
      float t0[8], t1[8], po = 0.f;
      #pragma unroll
      for (int r = 0; r < 8; ++r) {
        t0[r] = TANHF(d0[r]);
        t1[r] = TANHF(d1[r]);
        po = fmaf(t0[r], woA[r], po);
        po = fmaf(t1[r], woB[r], po);
      }
      const float pos = po + swapx16(po);   // combine lane n / n+16 halves (off-path)
      outv[u] = pos + bo;
      if (u == 3 && !up)
        *reinterpret_cast<v4f*>(&outrow[t4]) = outv;   // coalesced b128 every 4 steps

      // ---- zero-cost "transpose": D registers ARE the next B chunks (renaming) ----
      #pragma unroll
      for (int r = 0; r < 8; r += 2) {
        bc[r >> 1]       = (v2f){t0[r], t0[r + 1]};
        bc[4 + (r >> 1)] = (v2f){t1[r], t1[r + 1]};
      }
    }
  }

  // ---- epilogue: hT from the carried bc registers (contiguous unit pairs) ----
  float* hTrow = out + (size_t)Bsz * T + (size_t)b * HID;
  #pragma unroll
  for (int c = 0; c < 8; ++c) {
    const int off = (c < 4 ? 2 * c : 16 + 2 * (c - 4)) + h8;
    *reinterpret_cast<v2f*>(&hTrow[off]) = bc[c];
  }
}

extern "C" void kernel_launch(void* const* d_in, const int* in_sizes, int n_in,
                              void* d_out, int out_size, void* d_ws, size_t ws_size,
                              hipStream_t stream) {
  const float* x     = (const float*)d_in[0];
  const float* h0    = (const float*)d_in[1];
  const float* W_ih  = (const float*)d_in[2];
  const float* b_ih  = (const float*)d_in[3];
  const float* W_hh  = (const float*)d_in[4];
  const float* b_hh  = (const float*)d_in[5];
  const float* W_out = (const float*)d_in[6];
  const float* b_out = (const float*)d_in[7];
  float* out = (float*)d_out;

  const int Bsz = in_sizes[1] / HID;   // h0: [1,B,32]
  const int T   = in_sizes[0] / Bsz;   // x:  [B,T,1]; T multiple of 4 (512)

  dim3 grid(Bsz / 16), block(32);
  rnn_scan_kernel<<<grid, block, 0, stream>>>(x, h0, W_ih, b_ih, W_hh, b_hh,
                                              W_out, b_out, out, Bsz, T);
}